// GroupDense_22497038696908
// MI455X (gfx1250) — compile-verified
//
#include <hip/hip_runtime.h>

// Problem constants (from reference): x[8,2048,4096] fp32, kernel[64,64,64] fp32
#define BB 8
#define SS 2048
#define CC 4096
#define GG 64
#define UU 64
#define RR (BB * SS)          // 16384 flattened (b,s) rows
#define ROWS_PER_BLOCK 64
#define THREADS 128           // 4 waves of 32
#define LDS_STRIDE 68         // 64 + 4 pad -> conflict-free fragment reads

typedef float v2f __attribute__((ext_vector_type(2)));
typedef float v4f __attribute__((ext_vector_type(4)));
typedef float v8f __attribute__((ext_vector_type(8)));

__global__ __launch_bounds__(THREADS)
void group_dense_wmma(const float* __restrict__ x,
                      const float* __restrict__ w,
                      float* __restrict__ y) {
  __shared__ __align__(16) float Xs[ROWS_PER_BLOCK * LDS_STRIDE]; // 64 rows x 64 K (padded)
  __shared__ __align__(16) float Ws[UU * LDS_STRIDE];             // Ws[v][u] = w[g][u][v]

  const int g       = blockIdx.y;
  const int rowBase = blockIdx.x * ROWS_PER_BLOCK;
  const int tid     = threadIdx.x;

  // ---- Stage X tile: 64 rows x 64 cols of group g, float4-coalesced
  {
    const float* xg = x + (size_t)rowBase * CC + (size_t)g * UU;
#pragma unroll
    for (int j = 0; j < 8; ++j) {
      int idx = tid + THREADS * j;      // 0..1023 float4 slots
      int row = idx >> 4;               // 16 float4 per row
      int c4  = (idx & 15) << 2;
      v4f val = *(const v4f*)(xg + (size_t)row * CC + c4);
      *(v4f*)(&Xs[row * LDS_STRIDE + c4]) = val;
    }
  }
  // ---- Stage W_g transposed: Ws[v][u] (B-matrix wants per-lane contiguous K pairs)
  {
    const float* wg = w + (size_t)g * UU * UU;
#pragma unroll
    for (int j = 0; j < 32; ++j) {
      int idx = tid + THREADS * j;      // 0..4095, coalesced 128B per instruction
      int u = idx >> 6;
      int v = idx & 63;
      Ws[v * LDS_STRIDE + u] = wg[idx];
    }
  }
  __syncthreads();

  const int wave    = tid >> 5;         // 0..3 -> row tile 16*wave
  const int lane    = tid & 31;
  const int halfsel = lane >> 4;        // 0: K=0,1 ; 1: K=2,3 (32-bit A/B VGPR layout)
  const int l16     = lane & 15;

  // A fragment source: row (16*wave + l16), K base offset 2*halfsel
  const float* aBase = &Xs[(wave * 16 + l16) * LDS_STRIDE + 2 * halfsel];
  // B fragment source: Ws[v][u], v = 16*n + l16, K base offset 2*halfsel
  const float* bBase = &Ws[l16 * LDS_STRIDE + 2 * halfsel];

  v8f acc[4] = {};

#pragma unroll
  for (int t = 0; t < 16; ++t) {        // K = 64 in steps of 4
    v2f a = *(const v2f*)(aBase + 4 * t);
#pragma unroll
    for (int n = 0; n < 4; ++n) {       // 4 N-tiles of 16 -> full 64 outputs
      v2f b = *(const v2f*)(bBase + (size_t)(n * 16) * LDS_STRIDE + 4 * t);
      // (neg_a, A, neg_b, B, c_mod, C, reuse_a, reuse_b)
      acc[n] = __builtin_amdgcn_wmma_f32_16x16x4_f32(
          false, a, false, b, (short)0, acc[n], false, false);
    }
  }

  // ---- Epilogue: ReLU + wave-local LDS transpose + fully coalesced b128 stores.
  // Wave w owns Xs rows 16w..16w+15 exclusively (A-frags read only those rows;
  // same-wave DS ops are in-order), so no workgroup barrier is required.
  //
  // D layout: lane(0..15) = column 16n+l16 ; VGPR r = row (r + 8*halfsel) in tile.
  float* waveXs = &Xs[(wave * 16) * LDS_STRIDE];
#pragma unroll
  for (int n = 0; n < 4; ++n) {
#pragma unroll
    for (int r = 0; r < 8; ++r) {
      float v = acc[n][r];
      v = v > 0.0f ? v : 0.0f;
      int row = 8 * halfsel + r;                      // 0..15 within wave tile
      waveXs[row * LDS_STRIDE + n * 16 + l16] = v;    // conflict-free scatter
    }
  }
  // Read back row-major float4 and store: lanes 0-15 cover one full 256B row
  // segment, lanes 16-31 the next row -> every store is full-line coalesced.
  float* yg = y + (size_t)(rowBase + wave * 16) * CC + (size_t)g * UU;
#pragma unroll
  for (int j = 0; j < 8; ++j) {
    int idx = lane + 32 * j;            // 0..255 float4 slots in 16x64 tile
    int row = idx >> 4;
    int c4  = (idx & 15) << 2;
    v4f val = *(const v4f*)(&waveXs[row * LDS_STRIDE + c4]);
    *(v4f*)(yg + (size_t)row * CC + c4) = val;
  }
}

extern "C" void kernel_launch(void* const* d_in, const int* in_sizes, int n_in,
                              void* d_out, int out_size, void* d_ws, size_t ws_size,
                              hipStream_t stream) {
  const float* x = (const float*)d_in[0];   // [8, 2048, 4096] fp32
  const float* w = (const float*)d_in[1];   // [64, 64, 64] fp32
  float* y = (float*)d_out;                 // [8, 2048, 4096] fp32

  dim3 grid(RR / ROWS_PER_BLOCK, GG);       // 256 x 64 blocks
  dim3 block(THREADS);
  hipLaunchKernelGGL(group_dense_wmma, grid, block, 0, stream, x, w, y);
}